// GCN_9552007266490
// MI455X (gfx1250) — compile-verified
//
#include <hip/hip_runtime.h>
#include <hip/hip_bf16.h>

typedef float v2f __attribute__((ext_vector_type(2)));
typedef float v8f __attribute__((ext_vector_type(8)));

#define HF 128  // feature width (F == H == 128)

// ---------------- utility fills ----------------
__global__ __launch_bounds__(256) void fill_f32(float* p, float v, int n) {
    int i = blockIdx.x * 256 + threadIdx.x;
    if (i < n) p[i] = v;
}

// ---------------- degree / norm precompute ----------------
__global__ __launch_bounds__(256) void deg_scatter(const int* __restrict__ ei, float* deg, int E) {
    int e = blockIdx.x * 256 + threadIdx.x;
    if (e < E) atomicAdd(&deg[ei[E + e]], 1.0f);   // dst row of edge_index
}

__global__ __launch_bounds__(256) void make_dinv(const float* __restrict__ deg, float* dinv, int n) {
    int i = blockIdx.x * 256 + threadIdx.x;
    if (i < n) dinv[i] = rsqrtf(deg[i]);
}

__global__ __launch_bounds__(256) void make_norm(const int* __restrict__ ei, const float* __restrict__ dinv,
                                                 float* norm, int E) {
    int e = blockIdx.x * 256 + threadIdx.x;
    if (e < E) norm[e] = dinv[ei[e]] * dinv[ei[E + e]];
}

// ---------------- GEMM: H = X @ W  (N x 128 @ 128 x 128), f32 WMMA ----------------
// block = 256 threads = 8 waves; block computes a 16-row strip, wave w -> col tile w*16.
__global__ __launch_bounds__(256) void gemm_xw(const float* __restrict__ X, const float* __restrict__ W,
                                               float* __restrict__ Hout, int n) {
    __shared__ float sW[HF * HF];  // 64 KB: whole weight matrix
    int t = threadIdx.x;
    for (int i = t * 4; i < HF * HF; i += 256 * 4)
        *(float4*)(sW + i) = *(const float4*)(W + i);
    __syncthreads();

    int wave  = t >> 5;
    int lane  = t & 31;
    int row0  = blockIdx.x * 16;
    int col0  = wave * 16;
    int m     = lane & 15;        // row (A) / col (B,C) within tile
    int khalf = lane >> 4;        // 0: K pair {k,k+1}; 1: {k+2,k+3}

    int arow = row0 + m; if (arow >= n) arow = n - 1;   // clamp: keep EXEC all-ones for WMMA
    const float* ap = X + (long long)arow * HF;

    v8f c = {};
    for (int k0 = 0; k0 < HF; k0 += 4) {
        int kk = k0 + khalf * 2;
        v2f a = *(const v2f*)(ap + kk);                     // A[m][kk], A[m][kk+1]
        v2f b;
        b.x = sW[kk * HF + col0 + m];                       // B[kk][col]
        b.y = sW[(kk + 1) * HF + col0 + m];                 // B[kk+1][col]
        c = __builtin_amdgcn_wmma_f32_16x16x4_f32(false, a, false, b, (short)0, c, false, false);
    }

    float* op = Hout + (long long)row0 * HF + col0;
    #pragma unroll
    for (int v = 0; v < 8; ++v) {
        int mm = v + khalf * 8;                             // C: vgpr v -> row v / v+8
        if (row0 + mm < n) op[mm * HF + m] = c[v];
    }
}

// ---------------- edge scatter-add: agg[dst] += h[src] * norm ----------------
// one lane handles 4 features; 32 lanes cover 128 features of one edge.
__global__ __launch_bounds__(256) void scatter_edges(const float* __restrict__ Hn, const int* __restrict__ ei,
                                                     const float* __restrict__ norm, float* __restrict__ agg, int E) {
    int idx  = blockIdx.x * 256 + threadIdx.x;
    int e    = idx >> 5;
    if (e >= E) return;
    int lane = idx & 31;
    int s = ei[e], d = ei[E + e];
    float nm = norm[e];
    const float4 hv = *(const float4*)(Hn + (long long)s * HF + lane * 4);
    float* ap = agg + (long long)d * HF + lane * 4;
    atomicAdd(ap + 0, hv.x * nm);
    atomicAdd(ap + 1, hv.y * nm);
    atomicAdd(ap + 2, hv.z * nm);
    atomicAdd(ap + 3, hv.w * nm);
}

// ---------------- self-loop + bias + tanh (in place on agg) ----------------
__global__ __launch_bounds__(256) void finish_layer(float* __restrict__ agg, const float* __restrict__ h,
                                                    const float* __restrict__ dinv, const float* __restrict__ bias,
                                                    int n) {
    int idx = blockIdx.x * 256 + threadIdx.x;
    if (idx >= n * HF) return;
    int i = idx >> 7, f = idx & (HF - 1);
    float di = dinv[i];
    float v = agg[idx] + h[idx] * di * di + bias[f];
    agg[idx] = tanhf(v);
}

// ---------------- pooling ----------------
__device__ __forceinline__ unsigned enc_f32(float f) {
    unsigned b = __float_as_uint(f);
    return (b & 0x80000000u) ? ~b : (b | 0x80000000u);
}

__global__ __launch_bounds__(256) void pool_scatter(const float* __restrict__ h, const int* __restrict__ batch,
                                                    unsigned* gmaxu, float* gsum, float* cnt, int n) {
    int idx = blockIdx.x * 256 + threadIdx.x;
    if (idx >= n * HF) return;
    int i = idx >> 7, f = idx & (HF - 1);
    int g = batch[i];
    float v = h[idx];
    atomicMax(&gmaxu[g * HF + f], enc_f32(v));
    atomicAdd(&gsum[g * HF + f], v);
    if (f == 0) atomicAdd(&cnt[g], 1.0f);
}

// one block per graph g: build pooled[256], write pooled, then out = pooled @ Wout + bout
__global__ __launch_bounds__(256) void final_pool(const unsigned* __restrict__ gmaxu, const float* __restrict__ gsum,
                                                  const float* __restrict__ cnt, const float* __restrict__ Wout,
                                                  const float* __restrict__ bout, float* __restrict__ out,
                                                  int G, int C) {
    __shared__ float p[2 * HF];
    int g = blockIdx.x, t = threadIdx.x;
    float val;
    if (t < HF) {
        unsigned u = gmaxu[g * HF + t];
        unsigned b = (u & 0x80000000u) ? (u & 0x7fffffffu) : ~u;
        val = __uint_as_float(b);
    } else {
        float c = cnt[g]; c = c > 1.0f ? c : 1.0f;
        val = gsum[g * HF + (t - HF)] / c;
    }
    p[t] = val;
    out[G * C + g * 2 * HF + t] = val;   // pooled (second tuple element), after out [G,C]
    __syncthreads();
    if (t < C) {
        float acc = bout[t];
        #pragma unroll 8
        for (int j = 0; j < 2 * HF; ++j) acc += p[j] * Wout[j * C + t];
        out[g * C + t] = acc;
    }
}

// ---------------- host launch ----------------
extern "C" void kernel_launch(void* const* d_in, const int* in_sizes, int n_in,
                              void* d_out, int out_size, void* d_ws, size_t ws_size,
                              hipStream_t stream) {
    const float* x     = (const float*)d_in[0];
    const int*   ei    = (const int*)d_in[1];
    const int*   batch = (const int*)d_in[2];
    const float* Wl[4] = {(const float*)d_in[3], (const float*)d_in[5], (const float*)d_in[7], (const float*)d_in[9]};
    const float* bl[4] = {(const float*)d_in[4], (const float*)d_in[6], (const float*)d_in[8], (const float*)d_in[10]};
    const float* Wout  = (const float*)d_in[11];
    const float* bout  = (const float*)d_in[12];

    int n = in_sizes[0] / HF;          // 100000 nodes
    int E = in_sizes[1] / 2;           // 1.6M edges
    int C = in_sizes[12];              // 10 classes
    int G = out_size / (C + 2 * HF);   // 64 graphs

    // workspace carve-up
    char* ws = (char*)d_ws;
    size_t nodeBytes = (size_t)n * HF * sizeof(float);
    float*    bufA  = (float*)ws;             ws += nodeBytes;     // gemm output h
    float*    bufB  = (float*)ws;             ws += nodeBytes;     // agg / activation (layer input)
    float*    deg   = (float*)ws;             ws += (size_t)n * sizeof(float);
    float*    dinv  = (float*)ws;             ws += (size_t)n * sizeof(float);
    float*    norm  = (float*)ws;             ws += (size_t)E * sizeof(float);
    unsigned* gmaxu = (unsigned*)ws;          ws += (size_t)G * HF * sizeof(unsigned);
    float*    gsum  = (float*)ws;             ws += (size_t)G * HF * sizeof(float);
    float*    cnt   = (float*)ws;             ws += (size_t)G * sizeof(float);

    int gN   = (n + 255) / 256;
    int gNF  = (n * HF + 255) / 256;
    int gE   = (E + 255) / 256;
    int gE32 = (int)(((long long)E * 32 + 255) / 256);
    int gRow = (n + 15) / 16;

    // graph-invariant precompute (once, reused by all 4 layers)
    fill_f32<<<gN, 256, 0, stream>>>(deg, 1.0f, n);                 // self-loop degree
    deg_scatter<<<gE, 256, 0, stream>>>(ei, deg, E);
    make_dinv<<<gN, 256, 0, stream>>>(deg, dinv, n);
    make_norm<<<gE, 256, 0, stream>>>(ei, dinv, norm, E);

    // 4 GCN layers: gemm(in->A); zero(B); scatter(A->B); finish(B += A*dinv^2; tanh)
    const float* in = x;
    for (int l = 0; l < 4; ++l) {
        gemm_xw<<<gRow, 256, 0, stream>>>(in, Wl[l], bufA, n);
        fill_f32<<<gNF, 256, 0, stream>>>(bufB, 0.0f, n * HF);
        scatter_edges<<<gE32, 256, 0, stream>>>(bufA, ei, norm, bufB, E);
        finish_layer<<<gNF, 256, 0, stream>>>(bufB, bufA, dinv, bl[l], n);
        in = bufB;
    }

    // pooling (gmaxu init 0 == minimal key; gsum/cnt init 0) — contiguous fill
    int poolElems = G * HF * 2 + G;
    fill_f32<<<(poolElems + 255) / 256, 256, 0, stream>>>((float*)gmaxu, 0.0f, poolElems);
    pool_scatter<<<gNF, 256, 0, stream>>>(bufB, batch, gmaxu, gsum, cnt, n);
    final_pool<<<G, 256, 0, stream>>>(gmaxu, gsum, cnt, Wout, bout, (float*)d_out, G, C);
}